// AHD_Loss_Mine_65635690217854
// MI455X (gfx1250) — compile-verified
//
#include <hip/hip_runtime.h>

typedef __attribute__((ext_vector_type(2))) float v2f;
typedef __attribute__((ext_vector_type(4))) float v4f;
typedef __attribute__((ext_vector_type(8))) float v8f;

#define NROWS 8192
#define DDIM  64
#define NCHUNK 8
#define CHUNK (NROWS / NCHUNK)        // 1024 B-rows per block
#define TILES_PER_CHUNK (CHUNK / 16)  // 64 tiles of 16
#define ROWS_PER_BLOCK 128            // 8 waves x 16 rows
#define INF_BITS 0x7f800000u
#define USE_ASYNC 1                   // gfx1250 async-to-LDS staging path

// ---------------------------------------------------------------------------
// Init: squared norms + seed min arrays with +inf
// ---------------------------------------------------------------------------
__global__ void init_kernel(const float* __restrict__ X, const float* __restrict__ Y,
                            float* __restrict__ x2, float* __restrict__ y2,
                            unsigned* __restrict__ rowmin, unsigned* __restrict__ colmin) {
    int i = blockIdx.x * blockDim.x + threadIdx.x;
    if (i < NROWS) {
        const float* p = X + (size_t)i * DDIM;
        float s = 0.f;
        #pragma unroll
        for (int k = 0; k < DDIM; ++k) s = fmaf(p[k], p[k], s);
        x2[i] = s;
        rowmin[i] = INF_BITS;
    } else if (i < 2 * NROWS) {
        int j = i - NROWS;
        const float* p = Y + (size_t)j * DDIM;
        float s = 0.f;
        #pragma unroll
        for (int k = 0; k < DDIM; ++k) s = fmaf(p[k], p[k], s);
        y2[j] = s;
        colmin[j] = INF_BITS;
    }
}

// ---------------------------------------------------------------------------
// Fused GEMM + row-min kernel.
// Block = 8 waves; all waves share one B column-chunk, each wave owns a
// 16-row A tile.  B tiles (16x64 f32) are staged once per block into an LDS
// double buffer via global_load_async_to_lds_b64 (ASYNCcnt), pre-swizzled
// into WMMA B-fragment order: slot [s*32 + lane] holds
// B[colBase + (lane&15)][4s + 2*(lane>>4) + {0,1}].  Wave fragment reads are
// ds_load_b64 with 8-byte lane stride -> conflict-free across 64 banks.
// ---------------------------------------------------------------------------
__global__ __launch_bounds__(256) void chamfer_tile(
        const float* __restrict__ A,   // rows we reduce over
        const float* __restrict__ B,   // candidate set
        const float* __restrict__ b2,  // |B_j|^2
        const float* __restrict__ a2,  // |A_i|^2
        unsigned* __restrict__ outmin) // min d^2 (clamped >= 0) as uint bits
{
    __shared__ v2f bufs[2][512];       // 2 x 4KB double buffer

    const int t    = threadIdx.x;
    const int lane = t & 31;
    const int wave = t >> 5;
    const int chunk   = blockIdx.x % NCHUNK;
    const int rowBase = (blockIdx.x / NCHUNK) * ROWS_PER_BLOCK + wave * 16;
    const int lo = lane & 15;
    const int hi = lane >> 4;
    const int colBase0 = chunk * CHUNK;

    // --- A fragment, resident for the whole chunk loop (ISA 16x4 f32 layout)
    v2f xa[16];
    const float* arow = A + (size_t)(rowBase + lo) * DDIM + 2 * hi;
    #pragma unroll
    for (int s = 0; s < 16; ++s)
        xa[s] = *(const v2f*)(arow + 4 * s);

    // --- staging: thread t copies B[colBase + (t&15)][4*(t>>4) .. +3] (16B)
    const int srow = t & 15;
    const int sj   = t >> 4;
    const float*   gB0     = B + (size_t)(colBase0 + srow) * DDIM + 4 * sj;
    const unsigned ldsOff0 = (unsigned)(uintptr_t)(&bufs[0][0])
                           + (unsigned)((sj * 32 + srow) * 8);

    auto stage = [&](int ct) {
        const float* gp = gB0 + (size_t)ct * 16 * DDIM;
        unsigned l1 = ldsOff0 + (unsigned)((ct & 1) * 4096);
#if USE_ASYNC
        unsigned long long g = (unsigned long long)(uintptr_t)gp;
        unsigned l2 = l1 + 120;  // +8 from offset:8 lands at l1+128 (hi half)
        asm volatile("global_load_async_to_lds_b64 %0, %1, off"
                     :: "v"(l1), "v"(g) : "memory");
        asm volatile("global_load_async_to_lds_b64 %0, %1, off offset:8"
                     :: "v"(l2), "v"(g) : "memory");
#else
        v4f d = *(const v4f*)gp;
        v2f* dst = &bufs[ct & 1][sj * 32 + srow];
        dst[0]  = (v2f){d.x, d.y};   // lanes 0-15 slot
        dst[16] = (v2f){d.z, d.w};   // lanes 16-31 slot
#endif
    };

    stage(0);

    v8f runmin = 3.0e38f;  // splat
    const float* b2c = b2 + colBase0;

    for (int ct = 0; ct < TILES_PER_CHUNK; ++ct) {
        if (ct + 1 < TILES_PER_CHUNK) {
            stage(ct + 1);
#if USE_ASYNC
            asm volatile("s_wait_asynccnt 0x2" ::: "memory");  // tile ct done
#endif
        } else {
#if USE_ASYNC
            asm volatile("s_wait_asynccnt 0x0" ::: "memory");
#endif
        }
        __syncthreads();   // buf[ct&1] fully staged by all threads

        // Batch ALL fragment loads (8x ds_load_2addr_b64) + the b2 load, then
        // fence so the scheduler can't sink them into the WMMA chain.  One
        // s_wait_dscnt, then 16 back-to-back WMMAs.
        const v2f* buf = &bufs[ct & 1][0];
        v2f bb[16];
        #pragma unroll
        for (int s = 0; s < 16; ++s)
            bb[s] = buf[s * 32 + lane];          // conflict-free ds_load_b64
        const float bn = b2c[ct * 16 + lo];      // column n = lane%16
        asm volatile("" ::: "memory");           // scheduling fence

        v8f acc = {};
        #pragma unroll
        for (int s = 0; s < 16; ++s)
            acc = __builtin_amdgcn_wmma_f32_16x16x4_f32(
                false, xa[s], false, bb[s], (short)0, acc, false, false);

        #pragma unroll
        for (int r = 0; r < 8; ++r)
            runmin[r] = fminf(runmin[r], fmaf(-2.0f, acc[r], bn));

        __syncthreads();   // all reads of buf[ct&1] done before restage
    }

    // Min across the 16 lanes holding different columns of the same rows.
    #pragma unroll
    for (int off = 1; off < 16; off <<= 1) {
        #pragma unroll
        for (int r = 0; r < 8; ++r) {
            float o = __shfl_xor(runmin[r], off, 32);
            runmin[r] = fminf(runmin[r], o);
        }
    }

    if (lo == 0) {   // lane 0 -> rows rowBase+0..7 ; lane 16 -> +8..15
        #pragma unroll
        for (int r = 0; r < 8; ++r) {
            const int row = rowBase + 8 * hi + r;
            float d2 = fmaxf(runmin[r] + a2[row], 0.0f);  // >=0 => uint order
            atomicMin(&outmin[row], __float_as_uint(d2));
        }
    }
}

// ---------------------------------------------------------------------------
// Final reduction: mean(sqrt(rowmin)) + mean(sqrt(colmin)); N == M.
// ---------------------------------------------------------------------------
__global__ void reduce_kernel(const unsigned* __restrict__ rowmin,
                              const unsigned* __restrict__ colmin,
                              float* __restrict__ out) {
    __shared__ float sdata[256];
    const int t = threadIdx.x;
    float s = 0.f;
    for (int i = t; i < NROWS; i += 256) {
        s += sqrtf(__uint_as_float(rowmin[i]));
        s += sqrtf(__uint_as_float(colmin[i]));
    }
    sdata[t] = s;
    __syncthreads();
    for (int off = 128; off > 0; off >>= 1) {
        if (t < off) sdata[t] += sdata[t + off];
        __syncthreads();
    }
    if (t == 0) out[0] = sdata[0] * (1.0f / (float)NROWS);
}

// ---------------------------------------------------------------------------
extern "C" void kernel_launch(void* const* d_in, const int* in_sizes, int n_in,
                              void* d_out, int out_size, void* d_ws, size_t ws_size,
                              hipStream_t stream) {
    const float* X = (const float*)d_in[0];
    const float* Y = (const float*)d_in[1];

    float*    x2     = (float*)d_ws;
    float*    y2     = x2 + NROWS;
    unsigned* rowmin = (unsigned*)(y2 + NROWS);
    unsigned* colmin = rowmin + NROWS;

    init_kernel<<<(2 * NROWS + 255) / 256, 256, 0, stream>>>(X, Y, x2, y2, rowmin, colmin);

    const int blocks = (NROWS / ROWS_PER_BLOCK) * NCHUNK;   // 512

    // dis_1: for each X row, min over Y
    chamfer_tile<<<blocks, 256, 0, stream>>>(X, Y, y2, x2, rowmin);
    // dis_2: for each Y row, min over X (symmetric swap)
    chamfer_tile<<<blocks, 256, 0, stream>>>(Y, X, x2, y2, colmin);

    reduce_kernel<<<1, 256, 0, stream>>>(rowmin, colmin, (float*)d_out);
}